// GATModel_74844100100655
// MI455X (gfx1250) — compile-verified
//
#include <hip/hip_runtime.h>
#include <hip/hip_bf16.h>
#include <math.h>

#define BB 4
#define NB 8192
#define KN 16
#define MAXINE 32
#define ENCD 64
#define BIGF 1.0e9f
#define BN_TOTAL 32768       // B*N
#define NK 131072            // N*K
#define FLOWSTRIDE 131328    // NK + pad (zero sentinel lives at index NK)

typedef float v2f __attribute__((ext_vector_type(2)));
typedef float v8f __attribute__((ext_vector_type(8)));

// one K=4 step of a fp32 16x16 WMMA tile
__device__ __forceinline__ v8f wmma4(float ax, float ay, float bx, float by, v8f acc) {
    v2f a; a.x = ax; a.y = ay;
    v2f b; b.x = bx; b.y = by;
    return __builtin_amdgcn_wmma_f32_16x16x4_f32(false, a, false, b, (short)0, acc, false, false);
}

// ---------------- embedding max_norm=1 clip ----------------
__global__ __launch_bounds__(128) void k_embclip(const float* __restrict__ emb,
                                                 float* __restrict__ emb_c) {
    int row  = blockIdx.x * 4 + (threadIdx.x >> 5);
    int lane = threadIdx.x & 31;
    if (row >= NB) return;
    float a = emb[row * 64 + lane];
    float b = emb[row * 64 + 32 + lane];
    float ss = a * a + b * b;
    for (int m = 16; m >= 1; m >>= 1) ss += __shfl_xor(ss, m, 32);
    float nrm = sqrtf(ss);
    float scale = fminf(1.0f, 1.0f / fmaxf(nrm, 1e-12f));
    emb_c[row * 64 + lane]      = a * scale;
    emb_c[row * 64 + 32 + lane] = b * scale;
}

// ---------------- encoder MLP: tanh(tanh(x@W1+b1)@W2+b2) ----------------
__global__ __launch_bounds__(128) void k_encoder(const float* __restrict__ emb_c,
                                                 const float* __restrict__ feats,
                                                 const float* __restrict__ w1,
                                                 const float* __restrict__ b1,
                                                 const float* __restrict__ w2,
                                                 const float* __restrict__ b2,
                                                 float* __restrict__ enc) {
    __shared__ float xs[16][68];
    __shared__ float w1s[68][64];
    __shared__ float w2s[64][64];
    __shared__ float hs[16][64];
    int tid = threadIdx.x;
    int m0  = blockIdx.x * 16;
    for (int i = tid; i < 68 * 64; i += 128) {
        int r = i >> 6, c = i & 63;
        w1s[r][c] = (r < 66) ? w1[r * 64 + c] : 0.0f;
    }
    for (int i = tid; i < 64 * 64; i += 128) w2s[i >> 6][i & 63] = w2[i];
    for (int i = tid; i < 16 * 68; i += 128) {
        int r = i / 68, c = i % 68;
        int g = m0 + r;
        int n = g & (NB - 1);
        float v = 0.0f;
        if (c < 64)      v = emb_c[n * 64 + c];
        else if (c < 66) v = feats[(size_t)g * 2 + (c - 64)];
        xs[r][c] = v;
    }
    __syncthreads();
    int wv = tid >> 5, lane = tid & 31, hf = lane >> 4, l15 = lane & 15;
    int n0 = wv * 16;
    v8f acc = {};
    for (int k = 0; k < 68; k += 4)
        acc = wmma4(xs[l15][k + 2 * hf], xs[l15][k + 2 * hf + 1],
                    w1s[k + 2 * hf][n0 + l15], w1s[k + 2 * hf + 1][n0 + l15], acc);
    float bias = b1[n0 + l15];
#pragma unroll
    for (int i = 0; i < 8; i++) hs[i + 8 * hf][n0 + l15] = tanhf(acc[i] + bias);
    __syncthreads();
    v8f acc2 = {};
    for (int k = 0; k < 64; k += 4)
        acc2 = wmma4(hs[l15][k + 2 * hf], hs[l15][k + 2 * hf + 1],
                     w2s[k + 2 * hf][n0 + l15], w2s[k + 2 * hf + 1][n0 + l15], acc2);
    float bias2 = b2[n0 + l15];
#pragma unroll
    for (int i = 0; i < 8; i++) {
        int m = i + 8 * hf;
        enc[(size_t)(m0 + m) * 64 + n0 + l15] = tanhf(acc2[i] + bias2);
    }
}

// ---------------- GAT per-head transform: h = enc @ gat_w[h] ----------------
__global__ __launch_bounds__(128) void k_gat_h(const float* __restrict__ enc,
                                               const float* __restrict__ gw,
                                               float* __restrict__ hbuf) {
    __shared__ float es[16][64];
    __shared__ float wsm[64][64];
    int tid = threadIdx.x;
    int m0  = blockIdx.x * 16;
    for (int i = tid; i < 16 * 64; i += 128) es[i >> 6][i & 63] = enc[(size_t)m0 * 64 + i];
    int wv = tid >> 5, lane = tid & 31, hf = lane >> 4, l15 = lane & 15;
    int n0 = wv * 16;
    for (int hd = 0; hd < 4; hd++) {
        __syncthreads();
        for (int i = tid; i < 64 * 64; i += 128) wsm[i >> 6][i & 63] = gw[hd * 4096 + i];
        __syncthreads();
        v8f acc = {};
        for (int k = 0; k < 64; k += 4)
            acc = wmma4(es[l15][k + 2 * hf], es[l15][k + 2 * hf + 1],
                        wsm[k + 2 * hf][n0 + l15], wsm[k + 2 * hf + 1][n0 + l15], acc);
#pragma unroll
        for (int i = 0; i < 8; i++) {
            int m = i + 8 * hf;
            hbuf[((size_t)(m0 + m) * 4 + hd) * 64 + n0 + l15] = acc[i];
        }
    }
}

// ---------------- attention scores: s = h . a ----------------
__global__ __launch_bounds__(256) void k_scores(const float* __restrict__ hbuf,
                                                const float* __restrict__ a_self,
                                                const float* __restrict__ a_nbr,
                                                float* __restrict__ s_self,
                                                float* __restrict__ s_nbr) {
    int row  = blockIdx.x * 8 + (threadIdx.x >> 5);   // row = bn*4 + head
    int lane = threadIdx.x & 31;
    if (row >= BN_TOTAL * 4) return;
    int hd = row & 3;
    float h0 = hbuf[(size_t)row * 64 + lane];
    float h1 = hbuf[(size_t)row * 64 + 32 + lane];
    float ds = h0 * a_self[hd * 64 + lane] + h1 * a_self[hd * 64 + 32 + lane];
    float dn = h0 * a_nbr[hd * 64 + lane]  + h1 * a_nbr[hd * 64 + 32 + lane];
    for (int m = 16; m >= 1; m >>= 1) {
        ds += __shfl_xor(ds, m, 32);
        dn += __shfl_xor(dn, m, 32);
    }
    if (lane == 0) { s_self[row] = ds; s_nbr[row] = dn; }
}

// ---------------- attention softmax + neighbor aggregation ----------------
__global__ __launch_bounds__(256) void k_gat_att(const float* __restrict__ hbuf,
                                                 const float* __restrict__ s_self,
                                                 const float* __restrict__ s_nbr,
                                                 const int* __restrict__ adj,
                                                 const int* __restrict__ num_nodes,
                                                 float* __restrict__ enc_out) {
    __shared__ int   adj_s[8][16];
    __shared__ float alpha_s[8][16][4];
    int wid  = threadIdx.x >> 5;
    int lane = threadIdx.x & 31;
    int bn   = blockIdx.x * 8 + wid;
    int b    = bn >> 13;
    int k    = lane & 15;
    int nn   = num_nodes[b];
    int adjv = adj[(size_t)bn * 16 + k];
    float maskf = (adjv == nn) ? 1.0f : 0.0f;
    int nbn = b * NB + adjv;
    float al[4];
#pragma unroll
    for (int hd = 0; hd < 4; hd++) {
        float e = s_self[bn * 4 + hd] + s_nbr[(size_t)nbn * 4 + hd];
        e = (e >= 0.0f) ? e : 0.2f * e;      // leaky_relu(0.2)
        al[hd] = e - BIGF * maskf;
    }
#pragma unroll
    for (int hd = 0; hd < 4; hd++) {
        float mx = al[hd];
        for (int m = 8; m >= 1; m >>= 1) mx = fmaxf(mx, __shfl_xor(mx, m, 32));
        float ex = expf(al[hd] - mx);
        float sm = ex;
        for (int m = 8; m >= 1; m >>= 1) sm += __shfl_xor(sm, m, 32);
        al[hd] = ex / sm;
    }
    if (lane < 16) {
        adj_s[wid][k] = adjv;
#pragma unroll
        for (int hd = 0; hd < 4; hd++) alpha_s[wid][k][hd] = al[hd];
    }
    __syncthreads();
    int myhead = lane >> 3;
    int j0     = lane * 8;
    float acc[8];
#pragma unroll
    for (int t = 0; t < 8; t++) acc[t] = 0.0f;
    for (int kk = 0; kk < 16; kk++) {
        int nb   = adj_s[wid][kk];
        float av = alpha_s[wid][kk][myhead];
        const float4* p = reinterpret_cast<const float4*>(hbuf + (size_t)(b * NB + nb) * 256 + j0);
        float4 v0 = p[0], v1 = p[1];
        acc[0] += av * v0.x; acc[1] += av * v0.y; acc[2] += av * v0.z; acc[3] += av * v0.w;
        acc[4] += av * v1.x; acc[5] += av * v1.y; acc[6] += av * v1.z; acc[7] += av * v1.w;
    }
#pragma unroll
    for (int t = 0; t < 8; t++) {
        acc[t] += __shfl_xor(acc[t], 8, 32);
        acc[t] += __shfl_xor(acc[t], 16, 32);
    }
    if (lane < 8) {
#pragma unroll
        for (int t = 0; t < 8; t++)
            enc_out[(size_t)bn * 64 + lane * 8 + t] = tanhf(acc[t] * 0.25f);
    }
}

// ---------------- decoders: node_weights, dual_vars; also supply ----------------
__global__ __launch_bounds__(128) void k_decode(const float* __restrict__ enc,
                                                const float* __restrict__ dw1, const float* __restrict__ db1,
                                                const float* __restrict__ dw2, const float* __restrict__ db2,
                                                const float* __restrict__ uw1, const float* __restrict__ ub1,
                                                const float* __restrict__ uw2, const float* __restrict__ ub2,
                                                const float* __restrict__ dem,
                                                float* __restrict__ node_w,
                                                float* __restrict__ dual_v,
                                                float* __restrict__ supply) {
    __shared__ float es[16][64];
    __shared__ float wsm[64][64];
    __shared__ float hs[16][64];
    int tid = threadIdx.x;
    int m0  = blockIdx.x * 16;
    for (int i = tid; i < 16 * 64; i += 128) es[i >> 6][i & 63] = enc[(size_t)m0 * 64 + i];
    int wv = tid >> 5, lane = tid & 31, hf = lane >> 4, l15 = lane & 15;
    int n0 = wv * 16;
    for (int pass = 0; pass < 2; pass++) {
        const float* W1 = pass ? uw1 : dw1;
        const float* B1 = pass ? ub1 : db1;
        const float* W2 = pass ? uw2 : dw2;
        const float* B2 = pass ? ub2 : db2;
        float* outp     = pass ? dual_v : node_w;
        __syncthreads();
        for (int i = tid; i < 4096; i += 128) wsm[i >> 6][i & 63] = W1[i];
        __syncthreads();
        v8f acc = {};
        for (int k = 0; k < 64; k += 4)
            acc = wmma4(es[l15][k + 2 * hf], es[l15][k + 2 * hf + 1],
                        wsm[k + 2 * hf][n0 + l15], wsm[k + 2 * hf + 1][n0 + l15], acc);
        float bias = B1[n0 + l15];
#pragma unroll
        for (int i = 0; i < 8; i++) hs[i + 8 * hf][n0 + l15] = tanhf(acc[i] + bias);
        __syncthreads();
        int r = tid >> 3, p = tid & 7;      // 16 rows x 8 threads
        float partial = 0.0f;
        for (int c = p * 8; c < p * 8 + 8; c++) partial += hs[r][c] * W2[c];
        partial += __shfl_xor(partial, 1, 32);
        partial += __shfl_xor(partial, 2, 32);
        partial += __shfl_xor(partial, 4, 32);
        if (p == 0) outp[m0 + r] = partial + B2[0];
    }
    if (tid < 16) supply[m0 + tid] = fmaxf(-dem[m0 + tid], 0.0f);
}

// ---------------- edge-weight softmax + flow init ----------------
__global__ __launch_bounds__(256) void k_softmax_nw(const float* __restrict__ node_w,
                                                    const int* __restrict__ adj,
                                                    const int* __restrict__ num_nodes,
                                                    const float* __restrict__ supply,
                                                    float* __restrict__ nw_out,
                                                    float* __restrict__ flowA,
                                                    float* __restrict__ flowB) {
    int bn = blockIdx.x * 256 + threadIdx.x;
    if (bn >= BN_TOTAL) return;
    int b = bn >> 13, n = bn & (NB - 1);
    int nn = num_nodes[b];
    float w = node_w[bn];
    float pw[16];
    float mx = -3.4e38f;
#pragma unroll
    for (int k = 0; k < 16; k++) {
        int a = adj[(size_t)bn * 16 + k];
        pw[k] = (a == nn) ? (w - BIGF) : w;
        mx = fmaxf(mx, pw[k]);
    }
    float sm = 0.0f;
#pragma unroll
    for (int k = 0; k < 16; k++) { pw[k] = expf(pw[k] - mx); sm += pw[k]; }
    float inv = 1.0f / sm;
    float sup = supply[bn];
#pragma unroll
    for (int k = 0; k < 16; k++) {
        float a = pw[k] * inv;
        nw_out[(size_t)bn * 16 + k] = a;
        flowA[(size_t)b * FLOWSTRIDE + n * 16 + k] = a * sup;
    }
    if (n == 0) {                      // zero-pad sentinel at flattened index N*K
        flowA[(size_t)b * FLOWSTRIDE + NK] = 0.0f;
        flowB[(size_t)b * FLOWSTRIDE + NK] = 0.0f;
    }
}

// ---------------- one proportional-flow iteration ----------------
__global__ __launch_bounds__(256) void k_flow_step(const float* __restrict__ src,
                                                   const int* __restrict__ inidx,
                                                   const float* __restrict__ nw,
                                                   const float* __restrict__ supply,
                                                   float* __restrict__ dst,
                                                   int dstIsOut) {
    int bn = blockIdx.x * 256 + threadIdx.x;
    if (bn >= BN_TOTAL) return;
    int b = bn >> 13, n = bn & (NB - 1);
    const float* flat = src + (size_t)b * FLOWSTRIDE;
    float inflow = 0.0f;
#pragma unroll
    for (int j = 0; j < MAXINE; j++) inflow += flat[inidx[(size_t)bn * MAXINE + j]];
    float s = inflow + supply[bn];
    size_t base = dstIsOut ? ((size_t)bn * 16) : ((size_t)b * FLOWSTRIDE + n * 16);
#pragma unroll
    for (int k = 0; k < 16; k++) dst[base + k] = nw[(size_t)bn * 16 + k] * s;
}

// ---------------- flow cost partial reduction ----------------
__global__ __launch_bounds__(256) void k_flowcost(const float* __restrict__ flow,
                                                  float* __restrict__ partial) {
    int b = blockIdx.x >> 6, chunk = blockIdx.x & 63;
    const float* f = flow + (size_t)b * NK;
    float s = 0.0f;
    for (int i = chunk * 2048 + threadIdx.x; i < (chunk + 1) * 2048; i += 256) {
        float v = f[i];
        s += v * v;
    }
    __shared__ float red[256];
    red[threadIdx.x] = s;
    __syncthreads();
    for (int m = 128; m >= 1; m >>= 1) {
        if (threadIdx.x < m) red[threadIdx.x] += red[threadIdx.x + m];
        __syncthreads();
    }
    if (threadIdx.x == 0) partial[blockIdx.x] = red[0];
}

// ---------------- dual momentum descent (all 16 iters in registers) ----------------
__global__ __launch_bounds__(256) void k_dual(const float* __restrict__ dual_v,
                                              const int* __restrict__ adj,
                                              const int* __restrict__ num_nodes,
                                              const float* __restrict__ dem,
                                              float* __restrict__ partC,
                                              float* __restrict__ partD) {
    int bn = blockIdx.x * 256 + threadIdx.x;
    int b  = bn >> 13;
    int nn = num_nodes[b];
    float dv = dual_v[bn];
    float dd[16], f[16], mo[16], kp[16];
#pragma unroll
    for (int k = 0; k < 16; k++) {
        int a = adj[(size_t)bn * 16 + k];
        float maskf = (a == nn) ? 1.0f : 0.0f;
        kp[k] = 1.0f - maskf;
        float dtr = dual_v[b * NB + a] * kp[k];
        dd[k] = dtr - maskf * dv;
        f[k] = 0.0f;
        mo[k] = 0.0f;
    }
    for (int it = 0; it < 16; it++) {
#pragma unroll
        for (int k = 0; k < 16; k++) {
            float g = 2.0f * f[k] + dd[k];
            mo[k] = 0.9f * mo[k] + g;
            f[k] = fmaxf(f[k] - 0.01f * mo[k], 0.0f) * kp[k];
        }
    }
    float cost = 0.0f;
#pragma unroll
    for (int k = 0; k < 16; k++) cost += f[k] * f[k] + dd[k] * f[k];
    float dsum = dv * dem[bn];
    __shared__ float redC[256];
    __shared__ float redD[256];
    redC[threadIdx.x] = cost;
    redD[threadIdx.x] = dsum;
    __syncthreads();
    for (int m = 128; m >= 1; m >>= 1) {
        if (threadIdx.x < m) {
            redC[threadIdx.x] += redC[threadIdx.x + m];
            redD[threadIdx.x] += redD[threadIdx.x + m];
        }
        __syncthreads();
    }
    if (threadIdx.x == 0) { partC[blockIdx.x] = redC[0]; partD[blockIdx.x] = redD[0]; }
}

// ---------------- final scalar assembly ----------------
__global__ void k_final(const float* __restrict__ partFlow,
                        const float* __restrict__ partC,
                        const float* __restrict__ partD,
                        float* __restrict__ out) {
    int b = threadIdx.x;
    if (b >= BB) return;
    float fc = 0.0f;
    for (int i = 0; i < 64; i++) fc += partFlow[b * 64 + i];
    float c = 0.0f, d = 0.0f;
    for (int i = 0; i < 32; i++) { c += partC[b * 32 + i]; d += partD[b * 32 + i]; }
    float dual_cost = c - d;
    out[524292 + b]  = fc;          // flow_cost
    out[1048584 + b] = dual_cost;   // dual_cost
    out[b]           = fc - dual_cost; // loss
}

extern "C" void kernel_launch(void* const* d_in, const int* in_sizes, int n_in,
                              void* d_out, int out_size, void* d_ws, size_t ws_size,
                              hipStream_t stream) {
    (void)in_sizes; (void)n_in; (void)out_size; (void)ws_size;
    const float* feats   = (const float*)d_in[0];
    const float* dem     = (const float*)d_in[1];
    const int*   adj     = (const int*)d_in[2];
    const int*   inidx   = (const int*)d_in[3];
    const int*   nnodes  = (const int*)d_in[4];
    const float* emb     = (const float*)d_in[5];
    const float* enc_w1  = (const float*)d_in[6];
    const float* enc_b1  = (const float*)d_in[7];
    const float* enc_w2  = (const float*)d_in[8];
    const float* enc_b2  = (const float*)d_in[9];
    const float* gat_w   = (const float*)d_in[10];
    const float* a_self  = (const float*)d_in[11];
    const float* a_nbr   = (const float*)d_in[12];
    const float* dec_w1  = (const float*)d_in[13];
    const float* dec_b1  = (const float*)d_in[14];
    const float* dec_w2  = (const float*)d_in[15];
    const float* dec_b2  = (const float*)d_in[16];
    const float* dual_w1 = (const float*)d_in[17];
    const float* dual_b1 = (const float*)d_in[18];
    const float* dual_w2 = (const float*)d_in[19];
    const float* dual_b2 = (const float*)d_in[20];

    float* ws = (float*)d_ws;
    float* emb_c  = ws;                         // 524288
    float* encA   = ws + 524288;                // 2097152
    float* encB   = ws + 2621440;               // 2097152
    float* hbuf   = ws + 4718592;               // 8388608
    float* s_self = ws + 13107200;              // 131072
    float* s_nbr  = ws + 13238272;              // 131072
    float* node_w = ws + 13369344;              // 32768
    float* dual_v = ws + 13402112;              // 32768
    float* supply = ws + 13434880;              // 32768
    float* flowA  = ws + 13467648;              // 4*FLOWSTRIDE
    float* flowB  = ws + 13992960;              // 4*FLOWSTRIDE
    float* pFlow  = ws + 14518272;              // 256
    float* pC     = ws + 14518528;              // 128
    float* pD     = ws + 14518656;              // 128

    float* out      = (float*)d_out;
    float* out_flow = out + 4;
    float* out_nw   = out + 524296;

    k_embclip<<<2048, 128, 0, stream>>>(emb, emb_c);
    k_encoder<<<2048, 128, 0, stream>>>(emb_c, feats, enc_w1, enc_b1, enc_w2, enc_b2, encA);

    for (int layer = 0; layer < 2; layer++) {
        const float* ein = layer ? encB : encA;
        float* eout      = layer ? encA : encB;
        k_gat_h<<<2048, 128, 0, stream>>>(ein, gat_w, hbuf);
        k_scores<<<16384, 256, 0, stream>>>(hbuf, a_self, a_nbr, s_self, s_nbr);
        k_gat_att<<<4096, 256, 0, stream>>>(hbuf, s_self, s_nbr, adj, nnodes, eout);
    }

    k_decode<<<2048, 128, 0, stream>>>(encA, dec_w1, dec_b1, dec_w2, dec_b2,
                                       dual_w1, dual_b1, dual_w2, dual_b2,
                                       dem, node_w, dual_v, supply);
    k_softmax_nw<<<128, 256, 0, stream>>>(node_w, adj, nnodes, supply, out_nw, flowA, flowB);

    float* cur = flowA;
    float* nxt = flowB;
    for (int it = 0; it < 16; it++) {
        int last = (it == 15);
        float* dst = last ? out_flow : nxt;
        k_flow_step<<<128, 256, 0, stream>>>(cur, inidx, out_nw, supply, dst, last);
        float* t = cur; cur = nxt; nxt = t;
    }

    k_flowcost<<<256, 256, 0, stream>>>(out_flow, pFlow);
    k_dual<<<128, 256, 0, stream>>>(dual_v, adj, nnodes, dem, pC, pD);
    k_final<<<1, 32, 0, stream>>>(pFlow, pC, pD, out);
}